// Baseline_ResNet_LSTM_51539607552073
// MI455X (gfx1250) — compile-verified
//
#include <hip/hip_runtime.h>
#include <math.h>

// ---- model constants (match reference) ----
constexpr int kD    = 256;
constexpr int kHD   = 64;
constexpr int kNH   = 4;
constexpr int kFF   = 2048;
constexpr int kNL   = 3;
constexpr int kV    = 30000;
constexpr int kOLD  = 2048;
constexpr int kT    = 20;   // BPTT
constexpr int kB    = 64;   // batch
constexpr int kEOS  = 2;

typedef __attribute__((ext_vector_type(2))) float v2f;
typedef __attribute__((ext_vector_type(8))) float v8f;

// =====================================================================
// WMMA fp32 GEMM:  C[M,N] = act( A[M,K] @ W[K,N] + bias[N] (+ R[M,N]) )
// One wave32 computes a 16x(16*NB) super-tile; NB is a compile-time
// constant so the inner loop is branch-free straight-line code.
// All NB B fragments are loaded into distinct registers BEFORE the NB
// WMMAs so the scheduler can stagger s_wait_loadcnt and overlap matrix
// issue with outstanding loads (in-order load completion).
// Fragment layouts per ISA 7.12.2 (32-bit A 16x4 / B 4x16 / C 16x16).
// =====================================================================
template <int NB>
__global__ void gemm_wmma_f32(const float* __restrict__ A, const float* __restrict__ W,
                              const float* __restrict__ bias, const float* __restrict__ R,
                              float* __restrict__ C, int M, int N, int K, int relu,
                              int groupsN, int n_base)
{
    const int wavesPerBlock = blockDim.x >> 5;
    int wave = blockIdx.x * wavesPerBlock + (threadIdx.x >> 5);
    int lane = threadIdx.x & 31;
    int tilesM = M >> 4;
    int tM = wave / groupsN;
    int tG = wave - tM * groupsN;
    if (tM >= tilesM) return;                  // wave-uniform: EXEC stays all-ones
    int m0 = tM << 4;
    int n0 = n_base + tG * (NB * 16);
    int half = lane >> 4;                      // 0 = lanes 0-15, 1 = lanes 16-31
    int l16  = lane & 15;

    // A: lane covers row m0+l16; per k-step reads K(k+2*half), K(k+2*half+1)
    const float* Ap  = A + (size_t)(m0 + l16) * K + (half << 1);
    // B: lane covers col n0+l16(+16j); rows kk=k+2*half and kk+1
    const float* wr0 = W + (size_t)(half << 1) * N + n0 + l16;
    const size_t stepW = (size_t)4 * N;

    v8f acc[NB];
#pragma unroll
    for (int j = 0; j < NB; ++j) acc[j] = (v8f){};

#pragma unroll 2
    for (int k = 0; k < K; k += 4) {
        v2f a;
        a.x = Ap[0];
        a.y = Ap[1];
        // speculative prefetch of a future W row pair (global_prefetch_b8);
        // out-of-range speculative prefetches are dropped silently (ISA 10.5)
        __builtin_prefetch(wr0 + (size_t)16 * N, 0, 3);
        // issue ALL B-fragment loads first (distinct registers) ...
        v2f bf[NB];
#pragma unroll
        for (int j = 0; j < NB; ++j) {
            bf[j].x = wr0[j * 16];
            bf[j].y = wr0[j * 16 + N];
        }
        // ... then the NB WMMAs, so waits can stagger instead of draining
#pragma unroll
        for (int j = 0; j < NB; ++j) {
            acc[j] = __builtin_amdgcn_wmma_f32_16x16x4_f32(
                         /*neg_a=*/false, a, /*neg_b=*/false, bf[j],
                         /*c_mod=*/(short)0, acc[j],
                         /*reuse_a=*/false, /*reuse_b=*/false);
        }
        Ap  += 4;
        wr0 += stepW;
    }

#pragma unroll
    for (int j = 0; j < NB; ++j) {
        int n = n0 + j * 16 + l16;
        float bv = bias ? bias[n] : 0.0f;
#pragma unroll
        for (int r = 0; r < 8; ++r) {
            int m = m0 + (half << 3) + r;
            float v = acc[j][r] + bv;
            if (R)    v += R[(size_t)m * N + n];
            if (relu) v = v > 0.0f ? v : 0.0f;
            C[(size_t)m * N + n] = v;
        }
    }
}

// =====================================================================
// Small attention: one block (64 threads) per (batch b, head h). S <= 20.
// q[b,h,s,d] = qkv[(s*B+b)*768 + h*64 + d] ; k at +256 ; v at +512.
// out o[(s*B+b)*256 + h*64 + d]
// =====================================================================
__global__ void attn_kernel(const float* __restrict__ qkv, float* __restrict__ o, int S)
{
    __shared__ float sq[kT * kHD];
    __shared__ float sk[kT * kHD];
    __shared__ float sv[kT * kHD];
    __shared__ float sc[32];
    int b = blockIdx.x, h = blockIdx.y, d = threadIdx.x;   // d in [0,64)

    for (int s = 0; s < S; ++s) {
        size_t base = (size_t)(s * kB + b) * (3 * kD) + h * kHD + d;
        sq[s * kHD + d] = qkv[base];
        sk[s * kHD + d] = qkv[base + kD];
        sv[s * kHD + d] = qkv[base + 2 * kD];
    }
    __syncthreads();

    const float scale = 0.125f;   // 1/sqrt(64)
    for (int qi = 0; qi < S; ++qi) {
        if (d < S) {
            float acc = 0.0f;
            for (int dd = 0; dd < kHD; ++dd) acc += sq[qi * kHD + dd] * sk[d * kHD + dd];
            sc[d] = acc * scale;
        }
        __syncthreads();
        float m = -3.4e38f;
        for (int k = 0; k < S; ++k) m = fmaxf(m, sc[k]);
        float sum = 0.0f, out = 0.0f;
        for (int k = 0; k < S; ++k) {
            float e = expf(sc[k] - m);
            sum += e;
            out += e * sv[k * kHD + d];
        }
        o[(size_t)(qi * kB + b) * kD + h * kHD + d] = out / sum;
        __syncthreads();
    }
}

// =====================================================================
// LayerNorm over last dim (256). One block (256 threads) per row.
// =====================================================================
__global__ void layernorm_kernel(const float* __restrict__ x, const float* __restrict__ g,
                                 const float* __restrict__ be, float* __restrict__ y)
{
    __shared__ float red[kD];
    int row = blockIdx.x, t = threadIdx.x;
    float v = x[(size_t)row * kD + t];
    red[t] = v; __syncthreads();
    for (int s = kD / 2; s > 0; s >>= 1) { if (t < s) red[t] += red[t + s]; __syncthreads(); }
    float mu = red[0] * (1.0f / kD);
    __syncthreads();
    float dv = v - mu;
    red[t] = dv * dv; __syncthreads();
    for (int s = kD / 2; s > 0; s >>= 1) { if (t < s) red[t] += red[t + s]; __syncthreads(); }
    float var = red[0] * (1.0f / kD);
    y[(size_t)row * kD + t] = dv * rsqrtf(var + 1e-5f) * g[t] + be[t];
}

// =====================================================================
// mask_from_eos_after: cumprod of (tgt != EOS) with col 0 forced true.
// =====================================================================
__global__ void mask_kernel(const int* __restrict__ tgt, int* __restrict__ mask)
{
    int b = threadIdx.x;   // 64 threads
    int m = 1;
    for (int t = 0; t < kT; ++t) {
        int ne = (t == 0) ? 1 : ((tgt[b * kT + t] != kEOS) ? 1 : 0);
        m *= ne;
        mask[b * kT + t] = m;
    }
}

// =====================================================================
// Softmax over 30000 logits per row + argmax + running-max into ingr_probs
// + eos prob + masked sampled word. One block (256 threads) per batch row.
// =====================================================================
__global__ void softmax_update_kernel(const float* __restrict__ logits, const int* __restrict__ mask,
                                      int step, float* __restrict__ ingr, float* __restrict__ samp,
                                      float* __restrict__ eos, int* __restrict__ widx)
{
    __shared__ float sval[256];
    __shared__ int   sidx[256];
    int b = blockIdx.x, t = threadIdx.x;
    const float* row = logits + (size_t)b * kV;

    float m = -3.4e38f; int mi = kV;
    for (int v = t; v < kV; v += 256) {
        float x = row[v];
        if (x > m) { m = x; mi = v; }
    }
    sval[t] = m; sidx[t] = mi; __syncthreads();
    for (int s = 128; s > 0; s >>= 1) {
        if (t < s) {
            float ov = sval[t + s]; int oi = sidx[t + s];
            if (ov > sval[t] || (ov == sval[t] && oi < sidx[t])) { sval[t] = ov; sidx[t] = oi; }
        }
        __syncthreads();
    }
    float rmax = sval[0]; int rarg = sidx[0];
    __syncthreads();

    float s = 0.0f;
    for (int v = t; v < kV; v += 256) s += expf(row[v] - rmax);
    sval[t] = s; __syncthreads();
    for (int st = 128; st > 0; st >>= 1) { if (t < st) sval[t] += sval[t + st]; __syncthreads(); }
    float inv = 1.0f / sval[0];

    int mk = mask[b * kT + step];
    for (int v = t; v < kV; v += 256) {
        float p = expf(row[v] - rmax) * inv;
        if (v == kEOS) eos[b * kT + step] = p;
        if (mk) {
            float* ip = ingr + (size_t)b * kV + v;
            if (p > *ip) *ip = p;
        }
    }
    if (t == 0) {
        samp[b * kT + step] = mk ? (float)rarg : 0.0f;   // PAD = 0
        widx[b] = rarg;
    }
}

// embeds[step+1][b][:] = emb_table[widx[b]][:]
__global__ void embed_update_kernel(const int* __restrict__ widx, const float* __restrict__ emb,
                                    float* __restrict__ dst)
{
    int b = blockIdx.x, d = threadIdx.x;   // 64 blocks x 256 threads
    dst[b * kD + d] = emb[(size_t)widx[b] * kD + d];
}

// =====================================================================
static inline void launch_gemm(const float* A, const float* W, const float* bias, const float* R,
                               float* C, int M, int N, int K, int relu, hipStream_t s)
{
    int tilesM = M >> 4;
    int tilesN = N >> 4;
    int full   = tilesN >> 2;      // 64-wide groups
    int rem    = tilesN & 3;       // leftover 16-wide tiles (only logits: rem=3)
    if (full) {
        int waves  = tilesM * full;
        int blocks = (waves + 3) / 4;          // 4 waves (128 threads) / block
        gemm_wmma_f32<4><<<blocks, 128, 0, s>>>(A, W, bias, R, C, M, N, K, relu, full, 0);
    }
    if (rem) {
        int blocks = (tilesM + 3) / 4;         // one column group -> tilesM waves
        int n_base = full << 6;
        if (rem == 1)
            gemm_wmma_f32<1><<<blocks, 128, 0, s>>>(A, W, bias, R, C, M, N, K, relu, 1, n_base);
        else if (rem == 2)
            gemm_wmma_f32<2><<<blocks, 128, 0, s>>>(A, W, bias, R, C, M, N, K, relu, 1, n_base);
        else
            gemm_wmma_f32<3><<<blocks, 128, 0, s>>>(A, W, bias, R, C, M, N, K, relu, 1, n_base);
    }
}

extern "C" void kernel_launch(void* const* d_in, const int* in_sizes, int n_in,
                              void* d_out, int out_size, void* d_ws, size_t ws_size,
                              hipStream_t stream)
{
    (void)in_sizes; (void)n_in; (void)out_size; (void)ws_size;

    const float* image_feats = (const float*)d_in[0];
    const int*   target      = (const int*)  d_in[1];
    const float* emb_table   = (const float*)d_in[2];
    const float* W_g2e       = (const float*)d_in[3];
    const float* b_g2e       = (const float*)d_in[4];
    const float* W_out       = (const float*)d_in[5];
    const float* b_out       = (const float*)d_in[6];
    const float* Wqkv        = (const float*)d_in[7];
    const float* bqkv        = (const float*)d_in[8];
    const float* Wo          = (const float*)d_in[9];
    const float* bo          = (const float*)d_in[10];
    const float* W1          = (const float*)d_in[11];
    const float* b1          = (const float*)d_in[12];
    const float* W2          = (const float*)d_in[13];
    const float* b2          = (const float*)d_in[14];
    const float* g1          = (const float*)d_in[15];
    const float* be1         = (const float*)d_in[16];
    const float* g2          = (const float*)d_in[17];
    const float* be2         = (const float*)d_in[18];

    // outputs: [ingr_probs 64*30000][sampled_word 64*20][eos 64*20]
    float* out_ingr = (float*)d_out;
    float* out_samp = out_ingr + (size_t)kB * kV;
    float* out_eos  = out_samp + (size_t)kB * kT;

    // workspace carve-up (floats). logits aliases ff1 (never live together).
    float* ws     = (float*)d_ws;
    float* embeds = ws; ws += (size_t)kT * kB * kD;        // raw embeddings, row s*64+b
    float* xA     = ws; ws += (size_t)kT * kB * kD;        // encoder activations
    float* xB     = ws; ws += (size_t)kT * kB * kD;        // post-LN1 activations
    float* tmp    = ws; ws += (size_t)kT * kB * kD;        // residual sums
    float* qkvb   = ws; ws += (size_t)kT * kB * 3 * kD;    // qkv projections
    float* attno  = ws; ws += (size_t)kT * kB * kD;        // attention output
    float* ff1    = ws; ws += (size_t)kT * kB * kFF;       // FF hidden (2.62M floats)
    float* logits = ff1;                                   // 1.92M floats, aliased
    int*   mask   = (int*)ws;
    int*   widx   = mask + kB * kT;

    // ingr_probs is a running max of non-negative probs -> init to 0
    hipMemsetAsync(out_ingr, 0, (size_t)kB * kV * sizeof(float), stream);

    mask_kernel<<<1, kB, 0, stream>>>(target, mask);

    // img_embed = image_feats @ W_g2e + b_g2e  -> embeds rows 0..63 (position 0)
    launch_gemm(image_feats, W_g2e, b_g2e, nullptr, embeds, kB, kD, kOLD, 0, stream);

    for (int step = 0; step < kT; ++step) {
        int S = step + 1;
        int M = S * kB;

        // working copy of embeds[:S] (encoder mutates activations)
        hipMemcpyAsync(xA, embeds, (size_t)M * kD * sizeof(float),
                       hipMemcpyDeviceToDevice, stream);

        for (int l = 0; l < kNL; ++l) {
            const float* Wq  = Wqkv + (size_t)l * kD * 3 * kD;
            const float* bq  = bqkv + (size_t)l * 3 * kD;
            const float* Wol = Wo   + (size_t)l * kD * kD;
            const float* bol = bo   + (size_t)l * kD;
            const float* W1l = W1   + (size_t)l * kD * kFF;
            const float* b1l = b1   + (size_t)l * kFF;
            const float* W2l = W2   + (size_t)l * kFF * kD;
            const float* b2l = b2   + (size_t)l * kD;

            // qkv = x @ Wqkv + bqkv
            launch_gemm(xA, Wq, bq, nullptr, qkvb, M, 3 * kD, kD, 0, stream);
            // multi-head attention
            attn_kernel<<<dim3(kB, kNH), kHD, 0, stream>>>(qkvb, attno, S);
            // tmp = x + attn_o @ Wo + bo ; xB = LN(tmp, g1, be1)
            launch_gemm(attno, Wol, bol, xA, tmp, M, kD, kD, 0, stream);
            layernorm_kernel<<<M, kD, 0, stream>>>(tmp, g1 + (size_t)l * kD, be1 + (size_t)l * kD, xB);
            // ff1 = relu(xB @ W1 + b1) ; tmp = xB + ff1 @ W2 + b2 ; xA = LN(tmp, g2, be2)
            launch_gemm(xB, W1l, b1l, nullptr, ff1, M, kFF, kD, 1, stream);
            launch_gemm(ff1, W2l, b2l, xB, tmp, M, kD, kFF, 0, stream);
            layernorm_kernel<<<M, kD, 0, stream>>>(tmp, g2 + (size_t)l * kD, be2 + (size_t)l * kD, xA);
        }

        // logits = out[-1] @ W_out + b_out   (last position's 64 rows)
        launch_gemm(xA + (size_t)(S - 1) * kB * kD, W_out, b_out, nullptr,
                    logits, kB, kV, kD, 0, stream);

        softmax_update_kernel<<<kB, 256, 0, stream>>>(logits, mask, step,
                                                      out_ingr, out_samp, out_eos, widx);

        if (step + 1 < kT) {
            embed_update_kernel<<<kB, kD, 0, stream>>>(widx, emb_table,
                                                       embeds + (size_t)(step + 1) * kB * kD);
        }
    }
}